// Attention_41506563948971
// MI455X (gfx1250) — compile-verified
//
#include <hip/hip_runtime.h>
#include <math.h>

// ---------------------------------------------------------------------------
// Fused "transposed-softmax attention" for MI455X (gfx1250, wave32, WMMA).
//   P   = q @ w + b                       (kernel 1, WMMA f32 GEMM -> ws)
//   S   = a @ P^T ; softmax over q axis   (kernel 2, flash-style, 2 passes)
//   out = softmax(S) @ q
// fp32 throughout (precision requires it; compute-bound per roofline) via
// V_WMMA_F32_16X16X4_F32.  Staging uses GLOBAL_LOAD_ASYNC_TO_LDS_B128 with
// double-buffered LDS so HBM/L2 latency overlaps the matrix pipe.
// ---------------------------------------------------------------------------

typedef __attribute__((ext_vector_type(2))) float v2f;
typedef __attribute__((ext_vector_type(8))) float v8f;
typedef __attribute__((ext_vector_type(4))) int v4i;

#define BB   8
#define LL   2048
#define HH   1024
#define PSTR 1036  // LDS row stride (floats): 1036 % 64 = 12 -> 16 rows, 16 banks

// ---- async-to-LDS path (probe builtins; fall back to sync copies) ---------
#if defined(__has_builtin)
#if __has_builtin(__builtin_amdgcn_global_load_async_to_lds_b128) && \
    __has_builtin(__builtin_amdgcn_s_wait_asynccnt)
#define USE_ASYNC 1
#endif
#endif
#ifndef USE_ASYNC
#define USE_ASYNC 0
#endif

#define AS1 __attribute__((address_space(1)))
#define AS3 __attribute__((address_space(3)))

static __device__ __forceinline__ void async_cp16(const float* g, float* l) {
#if USE_ASYNC
  __builtin_amdgcn_global_load_async_to_lds_b128((AS1 v4i*)g, (AS3 v4i*)l,
                                                 /*offset=*/0, /*cpol=*/0);
#else
  *(float4*)l = *(const float4*)g;
#endif
}

#if USE_ASYNC
#define WAIT_ASYNC(n) __builtin_amdgcn_s_wait_asynccnt(n)
#else
#define WAIT_ASYNC(n) ((void)0)
#endif

static __device__ __forceinline__ v8f wmma_f32(v2f a, v2f b, v8f c) {
  // D = A(16x4) * B(4x16) + C(16x16), fp32
  return __builtin_amdgcn_wmma_f32_16x16x4_f32(false, a, false, b, (short)0, c,
                                               false, false);
}

// ---------------------------------------------------------------------------
// Kernel 1: P[b,i,j] = sum_h q[b,i,h] * w[h,j] + bias[j]
// 64x64 tile per 256-thread block (8 waves, each 16x32 of C).
// K staged by 16 with async double-buffered LDS tiles.
// ---------------------------------------------------------------------------
__global__ __launch_bounds__(256) void pgemm_kernel(
    const float* __restrict__ q, const float* __restrict__ w,
    const float* __restrict__ bias, float* __restrict__ P) {
  __shared__ float qt[2][64 * 20];  // 64 x 16 (+4 pad), double buffered
  __shared__ float wt[2][16 * 68];  // 16 x 64 (+4 pad), double buffered

  const int t = threadIdx.x;
  const int lane = t & 31, wv = t >> 5;
  const int l15 = lane & 15, lh = lane >> 4;
  const int m0 = (wv & 3) * 16;   // wave row offset in tile
  const int n0 = (wv >> 2) * 32;  // wave col offset in tile (2 N-tiles)
  const int mb = blockIdx.x * 64;
  const int nb = blockIdx.y * 64;
  const int b = blockIdx.z;
  const float* qB = q + (size_t)b * LL * HH;

  // Per-thread staging slots (2 async b128 per thread per stage).
  const int qrow = t >> 2, qc4 = (t & 3) * 4;
  const int wrow = t >> 4, wc4 = (t & 15) * 4;

  auto stage = [&](int k0, int buf) {
    async_cp16(qB + (size_t)(mb + qrow) * HH + k0 + qc4,
               &qt[buf][qrow * 20 + qc4]);
    async_cp16(w + (size_t)(k0 + wrow) * HH + nb + wc4,
               &wt[buf][wrow * 68 + wc4]);
  };

  v8f acc0 = {0.f, 0.f, 0.f, 0.f, 0.f, 0.f, 0.f, 0.f};
  v8f acc1 = acc0;

  stage(0, 0);
  int cur = 0;
  for (int k0 = 0; k0 < HH; k0 += 16) {
    const bool hasNext = (k0 + 16) < HH;
    if (hasNext) stage(k0 + 16, cur ^ 1);
    if (hasNext) WAIT_ASYNC(2); else WAIT_ASYNC(0);  // current tile landed
    __syncthreads();
#pragma unroll
    for (int kc = 0; kc < 4; ++kc) {
      const int ka = kc * 4 + 2 * lh;  // per-lane K base (A/B frag layout)
      v2f af = *(const v2f*)(&qt[cur][(m0 + l15) * 20 + ka]);
      v2f bf0, bf1;
      bf0.x = wt[cur][ka * 68 + n0 + l15];
      bf0.y = wt[cur][(ka + 1) * 68 + n0 + l15];
      bf1.x = wt[cur][ka * 68 + n0 + 16 + l15];
      bf1.y = wt[cur][(ka + 1) * 68 + n0 + 16 + l15];
      acc0 = wmma_f32(af, bf0, acc0);
      acc1 = wmma_f32(af, bf1, acc1);
    }
    __syncthreads();  // readers done before this buffer is re-staged
    cur ^= 1;
  }

  // bias add + store (C layout: VGPR j -> M = j + 8*laneHalf, N = lane&15)
#pragma unroll
  for (int tt = 0; tt < 2; ++tt) {
    union { v8f v; float f[8]; } u;
    u.v = tt ? acc1 : acc0;
    const int col = nb + n0 + tt * 16 + l15;
    const float bv = bias[col];
#pragma unroll
    for (int j = 0; j < 8; ++j) {
      const int row = mb + m0 + j + 8 * lh;
      P[(size_t)b * LL * HH + (size_t)row * HH + col] = u.f[j] + bv;
    }
  }
}

// ---------------------------------------------------------------------------
// Kernel 2: one (batch, 16-row a-block) per 256-thread block (8 waves).
// Wave w owns H-chunk [128w,128w+128): a-frags resident (64 VGPR), O in 64.
// Pass 1: online max/sum-exp stats.  Pass 2: O += exp(S-m) @ q.
// P blocks (and q blocks in pass 2) stream via async-to-LDS, double buffered.
// Cross-wave S reduction via ds_add_f32.
// LDS buffer pointers are computed from the toggle index (no pointer arrays
// of the extern-shared symbol: those become addrspacecast static
// initializers that ld.lld rejects).
// ---------------------------------------------------------------------------
#define PBUF(i) (smem + (size_t)(i) * (16 * PSTR))        // P block buffer i
#define QBUF(i) (smem + (size_t)(2 + (i)) * (16 * PSTR))  // q block buffer i

__global__ __launch_bounds__(256) void attn_kernel(
    const float* __restrict__ q, const float* __restrict__ a,
    const float* __restrict__ P, float* __restrict__ out) {
  extern __shared__ float smem[];
  float* Sbuf = smem + 4 * 16 * PSTR;  // 16x16 reduced S tile
  float* mLds = Sbuf + 256;            // 16 row maxes
  float* zLds = mLds + 16;             // 16 row sum-exp

  const int t = threadIdx.x, lane = t & 31, wv = t >> 5;
  const int l15 = lane & 15, lh = lane >> 4;
  const int b = blockIdx.x >> 7;
  const int ab = (blockIdx.x & 127) * 16;  // a-row block start
  const int hw = wv * 128;                 // wave's H chunk
  const float* qB = q + (size_t)b * LL * HH;
  const float* aB = a + (size_t)b * LL * HH;
  const float* PB = P + (size_t)b * LL * HH;

  // Resident A fragments: a[ab..ab+16) x H-chunk, in WMMA A layout.
  v2f afrag[32];
#pragma unroll
  for (int kc = 0; kc < 32; ++kc)
    afrag[kc] =
        *(const v2f*)(aB + (size_t)(ab + l15) * HH + hw + kc * 4 + 2 * lh);

  if (t < 16) { mLds[t] = -3.0e38f; zLds[t] = 0.f; }

  // Stage a 16x1024 block into an LDS buffer: 16 async b128 per thread.
  auto stage16 = [&](const float* src, float* dst) {
#pragma unroll
    for (int i = 0; i < 16; ++i) {
      const int g = i * 256 + t;
      const int row = g >> 8, c4 = (g & 255) * 4;
      async_cp16(src + (size_t)row * HH + c4, &dst[row * PSTR + c4]);
    }
  };

  // Partial S = a_blk @ Pblk^T over this wave's K=128, reduced into Sbuf.
  auto computeS = [&](const float* pb) {
    v8f s = {0.f, 0.f, 0.f, 0.f, 0.f, 0.f, 0.f, 0.f};
    const int base = l15 * PSTR + hw + 2 * lh;  // LDS row = q col n, col = h
#pragma unroll
    for (int kc = 0; kc < 32; ++kc) {
      v2f bf = *(const v2f*)(&pb[base + kc * 4]);
      s = wmma_f32(afrag[kc], bf, s);
    }
    union { v8f v; float f[8]; } u;
    u.v = s;
#pragma unroll
    for (int j = 0; j < 8; ++j)
      atomicAdd(&Sbuf[(j + 8 * lh) * 16 + l15], u.f[j]);
  };

  // ---------------- Pass 1: softmax statistics over q ----------------
  stage16(PB, PBUF(0));
  int cur = 0;
  for (int qb = 0; qb < LL; qb += 16) {
    const bool hasNext = (qb + 16) < LL;
    Sbuf[t] = 0.f;
    if (hasNext) stage16(PB + (size_t)(qb + 16) * HH, PBUF(cur ^ 1));
    if (hasNext) WAIT_ASYNC(16); else WAIT_ASYNC(0);
    __syncthreads();
    computeS(PBUF(cur));
    __syncthreads();
    if (t < 16) {  // online update of (m, Z) for a-row t
      float mblk = -3.0e38f;
      for (int c = 0; c < 16; ++c) mblk = fmaxf(mblk, Sbuf[t * 16 + c]);
      const float mo = mLds[t];
      const float mn = fmaxf(mo, mblk);
      float z = zLds[t] * __expf(mo - mn);
      for (int c = 0; c < 16; ++c) z += __expf(Sbuf[t * 16 + c] - mn);
      mLds[t] = mn;
      zLds[t] = z;
    }
    __syncthreads();
    cur ^= 1;
  }

  // ---------------- Pass 2: O += exp(S - m) @ q_blk ----------------
  v8f o[8];
#pragma unroll
  for (int i = 0; i < 8; ++i)
    o[i] = (v8f){0.f, 0.f, 0.f, 0.f, 0.f, 0.f, 0.f, 0.f};

  stage16(PB, PBUF(0));
  stage16(qB, QBUF(0));
  cur = 0;
  for (int qb = 0; qb < LL; qb += 16) {
    const bool hasNext = (qb + 16) < LL;
    Sbuf[t] = 0.f;
    if (hasNext) {
      stage16(PB + (size_t)(qb + 16) * HH, PBUF(cur ^ 1));
      stage16(qB + (size_t)(qb + 16) * HH, QBUF(cur ^ 1));
    }
    if (hasNext) WAIT_ASYNC(32); else WAIT_ASYNC(0);
    __syncthreads();
    computeS(PBUF(cur));
    __syncthreads();
    Sbuf[t] = __expf(Sbuf[t] - mLds[t >> 4]);  // E tile, row = t>>4
    __syncthreads();

    v2f ef[4];  // E as WMMA A operand (M = a rows, K = q cols)
#pragma unroll
    for (int kc = 0; kc < 4; ++kc)
      ef[kc] = *(const v2f*)(&Sbuf[l15 * 16 + kc * 4 + 2 * lh]);
    const float* qb_lds = QBUF(cur);
#pragma unroll
    for (int nt = 0; nt < 8; ++nt) {
#pragma unroll
      for (int kc = 0; kc < 4; ++kc) {
        const int k0 = kc * 4 + 2 * lh;
        const int col = hw + nt * 16 + l15;
        v2f bf;
        bf.x = qb_lds[k0 * PSTR + col];
        bf.y = qb_lds[(k0 + 1) * PSTR + col];
        o[nt] = wmma_f32(ef[kc], bf, o[nt]);
      }
    }
    __syncthreads();
    cur ^= 1;
  }

  // Normalize by Z and store out[b, ab+row, hw+col].
  float zinv[8];
#pragma unroll
  for (int j = 0; j < 8; ++j) zinv[j] = 1.0f / zLds[j + 8 * lh];
  float* oB = out + (size_t)b * LL * HH;
#pragma unroll
  for (int nt = 0; nt < 8; ++nt) {
    union { v8f v; float f[8]; } u;
    u.v = o[nt];
#pragma unroll
    for (int j = 0; j < 8; ++j) {
      const int row = ab + j + 8 * lh;
      const int col = hw + nt * 16 + l15;
      oB[(size_t)row * HH + col] = u.f[j] * zinv[j];
    }
  }
}

// ---------------------------------------------------------------------------
extern "C" void kernel_launch(void* const* d_in, const int* in_sizes, int n_in,
                              void* d_out, int out_size, void* d_ws,
                              size_t ws_size, hipStream_t stream) {
  const float* q = (const float*)d_in[0];
  const float* a = (const float*)d_in[1];
  const float* w = (const float*)d_in[2];
  const float* bias = (const float*)d_in[3];
  float* out = (float*)d_out;
  float* P = (float*)d_ws;  // B*L*H fp32 = 64 MB scratch

  const size_t shmem = (size_t)(4 * 16 * PSTR + 256 + 32) * sizeof(float);
  static bool attr_set = false;  // idempotent host-side attribute, not a stream op
  if (!attr_set) {
    (void)hipFuncSetAttribute((const void*)attn_kernel,
                              hipFuncAttributeMaxDynamicSharedMemorySize,
                              (int)shmem);
    attr_set = true;
  }

  dim3 g1(LL / 64, HH / 64, BB);
  pgemm_kernel<<<g1, 256, 0, stream>>>(q, w, bias, P);

  attn_kernel<<<dim3(BB * (LL / 16)), 256, shmem, stream>>>(q, a, P, out);
}